// NeuralODE_TVAR_88828513616171
// MI455X (gfx1250) — compile-verified
//
#include <hip/hip_runtime.h>

#define B_ 1024
#define T_ 4096
#define L_ 30
#define H_ 32
#define P_ 2

typedef __attribute__((ext_vector_type(2))) float v2f;
typedef __attribute__((ext_vector_type(8))) float v8f;

// CDNA5 hardware tanh (TRANS op). ISA requires 1 independent op / V_NOP after
// a TRANS before its result is consumed.
__device__ __forceinline__ float tanh_hw(float x) {
  float r;
  asm volatile("v_tanh_f32 %0, %1\n\tv_nop" : "=v"(r) : "v"(x));
  return r;
}

// ---------------------------------------------------------------------------
// Phase 1: encoder  z0 = tanh(X[:, :L] @ W1 + b1) @ W2 + b2   via WMMA f32.
// One wave handles 16 batch rows. K = 30 padded to 32 (B rows 30,31 zeroed).
// ---------------------------------------------------------------------------
__global__ __launch_bounds__(128) void encoder_kernel(
    const float* __restrict__ x_seq,   // (B, T)
    const float* __restrict__ enc_w1,  // (L, H)
    const float* __restrict__ enc_b1,  // (H)
    const float* __restrict__ enc_w2,  // (H, P)
    const float* __restrict__ enc_b2,  // (P)
    float* __restrict__ z_out)         // (B, T, P): writes z[:,0,:]
{
  const int tid  = blockIdx.x * blockDim.x + threadIdx.x;
  const int wave = tid >> 5;
  const int lane = tid & 31;
  const int half = lane >> 4;
  const int l    = lane & 15;
  const int rb   = wave * 16;   // first batch row of this wave's tile
  const int row  = rb + l;

  v8f acc0 = {};  // H columns 0..15
  v8f acc1 = {};  // H columns 16..31

#pragma unroll
  for (int kc = 0; kc < 8; ++kc) {
    const int k0 = kc * 4 + 2 * half;
    v2f a;
    a.x = x_seq[(size_t)row * T_ + k0];
    a.y = x_seq[(size_t)row * T_ + k0 + 1];
    v2f b0, b1v;
    b0.x  = (k0     < L_) ? enc_w1[(k0    ) * H_ + l     ] : 0.f;
    b0.y  = (k0 + 1 < L_) ? enc_w1[(k0 + 1) * H_ + l     ] : 0.f;
    b1v.x = (k0     < L_) ? enc_w1[(k0    ) * H_ + 16 + l] : 0.f;
    b1v.y = (k0 + 1 < L_) ? enc_w1[(k0 + 1) * H_ + 16 + l] : 0.f;
    acc0 = __builtin_amdgcn_wmma_f32_16x16x4_f32(false, a, false, b0,
                                                 (short)0, acc0, false, false);
    acc1 = __builtin_amdgcn_wmma_f32_16x16x4_f32(false, a, false, b1v,
                                                 (short)0, acc1, false, false);
  }

  const float bias0 = enc_b1[l];
  const float bias1 = enc_b1[16 + l];
  float h0[8], h1[8];
#pragma unroll
  for (int r = 0; r < 8; ++r) {
    h0[r] = tanh_hw(acc0[r] + bias0);
    h1[r] = tanh_hw(acc1[r] + bias1);
  }

  const float w20a = enc_w2[l * P_ + 0];
  const float w20b = enc_w2[l * P_ + 1];
  const float w21a = enc_w2[(16 + l) * P_ + 0];
  const float w21b = enc_w2[(16 + l) * P_ + 1];
  const float b2a = enc_b2[0], b2b = enc_b2[1];

#pragma unroll
  for (int r = 0; r < 8; ++r) {
    float s0 = h0[r] * w20a + h1[r] * w21a;
    float s1 = h0[r] * w20b + h1[r] * w21b;
#pragma unroll
    for (int m = 1; m <= 8; m <<= 1) {
      s0 += __shfl_xor(s0, m, 32);
      s1 += __shfl_xor(s1, m, 32);
    }
    if (l == 0) {
      const int brow = rb + r + 8 * half;
      v2f z0v; z0v.x = s0 + b2a; z0v.y = s1 + b2b;
      *(v2f*)(z_out + (size_t)brow * T_ * P_) = z0v;  // z[b][0][:]
    }
  }
}

// ---------------------------------------------------------------------------
// Phase 2: sequential Euler scan. 8 lanes per batch element (4 hidden units
// per lane), 4 elements per wave, 256 waves. In-lane tree over 4 products +
// 3-level butterfly (masks 1,2,4) => 3 cross-lane hops on the critical path
// instead of 5. Fuses kappa -> AR(2) -> teacher-forced prediction + stores.
// ---------------------------------------------------------------------------
__global__ __launch_bounds__(256) void scan_kernel(
    const float* __restrict__ phi,    // (B, T, P)
    const float* __restrict__ f_w1,   // (P, H)
    const float* __restrict__ f_b1,   // (H)
    const float* __restrict__ f_w2,   // (H, P)
    const float* __restrict__ f_b2,   // (P)
    float* __restrict__ xhat,         // (B, T)
    float* __restrict__ a_out,        // (B, T, P)
    float* __restrict__ z_out)        // (B, T, P), z[:,0,:] pre-filled
{
  const int tid  = blockIdx.x * blockDim.x + threadIdx.x;
  const int wave = tid >> 5;
  const int lane = tid & 31;
  const int grp  = lane >> 3;    // element slot within wave (0..3)
  const int gl   = lane & 7;     // lane within element group
  const int b    = wave * 4 + grp;

  // This lane's 4 hidden units: u .. u+3
  const int u = gl * 4;
  float w1a[4], w1b[4], bb1[4], w2a[4], w2b[4];
#pragma unroll
  for (int k = 0; k < 4; ++k) {
    w1a[k] = f_w1[u + k];              // f_w1[0][u+k]
    w1b[k] = f_w1[H_ + u + k];         // f_w1[1][u+k]
    bb1[k] = f_b1[u + k];
    w2a[k] = f_w2[(u + k) * P_ + 0];
    w2b[k] = f_w2[(u + k) * P_ + 1];
  }
  const float b2a = f_b2[0];
  const float b2b = f_b2[1];

  float*       zrow = z_out + (size_t)b * T_ * P_;
  float*       arow = a_out + (size_t)b * T_ * P_;
  float*       xrow = xhat  + (size_t)b * T_;
  const float* prow = phi   + (size_t)b * T_ * P_;

  float z1 = zrow[0];
  float z2 = zrow[1];

  // One Euler step: z += tanh(z @ W1 + b1) @ W2 + b2  (replicated in group)
  auto step = [&]() {
    float h[4];
#pragma unroll
    for (int k = 0; k < 4; ++k)
      h[k] = tanh_hw(fmaf(z2, w1b[k], fmaf(z1, w1a[k], bb1[k])));
    float s1 = (h[0] * w2a[0] + h[1] * w2a[1]) + (h[2] * w2a[2] + h[3] * w2a[3]);
    float s2 = (h[0] * w2b[0] + h[1] * w2b[1]) + (h[2] * w2b[2] + h[3] * w2b[3]);
#pragma unroll
    for (int m = 1; m <= 4; m <<= 1) {   // stays within the 8-lane group
      s1 += __shfl_xor(s1, m, 32);
      s2 += __shfl_xor(s2, m, 32);
    }
    z1 += s1 + b2a;
    z2 += s2 + b2b;
  };

  // t = 0 outputs (z[b][0] already written by encoder; xhat[:, :P] = 0)
  {
    const float k1 = tanh_hw(z1), k2 = tanh_hw(z2);
    if (gl == 0) {
      v2f av; av.x = k1 * (1.f - k2); av.y = k2;
      *(v2f*)arow = av;
      xrow[0] = 0.f;
    }
  }

  // t = 1 (peeled: xhat[b][1] = 0, no phi needed)
  step();
  {
    const float k1 = tanh_hw(z1), k2 = tanh_hw(z2);
    if (gl == 0) {
      v2f zv; zv.x = z1; zv.y = z2;
      *(v2f*)(zrow + P_) = zv;
      v2f av; av.x = k1 * (1.f - k2); av.y = k2;
      *(v2f*)(arow + P_) = av;
      xrow[1] = 0.f;
    }
  }

  // t = 2 .. T-1: branch-free body, single predicated store region
  for (int t = 2; t < T_; ++t) {
    const v2f ph = *(const v2f*)(prow + t * P_);  // same addr per group -> 1 req
    step();
    const float k1 = tanh_hw(z1);
    const float k2 = tanh_hw(z2);
    const float a1 = k1 * (1.f - k2);
    const float a2 = k2;
    const float xh = a1 * ph.x + a2 * ph.y;
    if (gl == 0) {
      v2f zv; zv.x = z1; zv.y = z2;
      *(v2f*)(zrow + t * P_) = zv;
      v2f av; av.x = a1; av.y = a2;
      *(v2f*)(arow + t * P_) = av;
      xrow[t] = xh;
    }
  }
}

extern "C" void kernel_launch(void* const* d_in, const int* in_sizes, int n_in,
                              void* d_out, int out_size, void* d_ws, size_t ws_size,
                              hipStream_t stream) {
  const float* x_seq  = (const float*)d_in[0];
  const float* phi    = (const float*)d_in[1];
  const float* enc_w1 = (const float*)d_in[2];
  const float* enc_b1 = (const float*)d_in[3];
  const float* enc_w2 = (const float*)d_in[4];
  const float* enc_b2 = (const float*)d_in[5];
  const float* f_w1   = (const float*)d_in[6];
  const float* f_b1   = (const float*)d_in[7];
  const float* f_w2   = (const float*)d_in[8];
  const float* f_b2   = (const float*)d_in[9];

  float* out  = (float*)d_out;
  float* xh   = out;                                   // (B, T)
  float* a_o  = out + (size_t)B_ * T_;                 // (B, T, P)
  float* z_o  = a_o + (size_t)B_ * T_ * P_;            // (B, T, P)

  // Phase 1: WMMA encoder fills z[:, 0, :]. 64 waves = B/16 row-tiles.
  encoder_kernel<<<16, 128, 0, stream>>>(x_seq, enc_w1, enc_b1, enc_w2, enc_b2, z_o);
  // Phase 2: 256 waves, 4 batch elements each, run the scan + fused outputs.
  scan_kernel<<<32, 256, 0, stream>>>(phi, f_w1, f_b1, f_w2, f_b2, xh, a_o, z_o);
}